// CoreModel_34110630265161
// MI455X (gfx1250) — compile-verified
//
#include <hip/hip_runtime.h>
#include <stdint.h>

// Problem constants (match reference): B=1, K=24, H=W=512, N=200000, C=13
#define KK   24
#define HH   512
#define WW   512
#define HWPX (HH * WW)          // 262144 pixels
#define NPTS 200000
#define CF   13
#define RECW 16                 // padded record width (floats): 13 feats + inv_r2 + pad
#define TPB  256

typedef unsigned int u32x4 __attribute__((ext_vector_type(4)));
typedef int          i32x4 __attribute__((ext_vector_type(4)));
typedef int          i32x8 __attribute__((ext_vector_type(8)));

#if __has_builtin(__builtin_amdgcn_tensor_load_to_lds) && \
    __has_builtin(__builtin_amdgcn_s_wait_tensorcnt)
#define USE_TDM 1
#else
#define USE_TDM 0
#endif

// ---------------------------------------------------------------------------
// Kernel 1: pack per-point data into 64B records: [feat0..feat12, 1/r^2, 0, 0]
// Turns 13 strided gathers + 1 radius gather per fragment into 4 aligned
// float4 loads of one 64B record.
// ---------------------------------------------------------------------------
__global__ __launch_bounds__(256)
void pack_points_kernel(const float* __restrict__ feat,    // [C, N]
                        const float* __restrict__ radius,  // [N]
                        float* __restrict__ rec)           // [N, 16]
{
    int n = blockIdx.x * blockDim.x + threadIdx.x;
    if (n >= NPTS) return;

    float v[RECW];
#pragma unroll
    for (int c = 0; c < CF; ++c)
        v[c] = __builtin_nontemporal_load(&feat[(size_t)c * NPTS + n]); // read-once stream
    float r  = radius[n];
    v[13] = 1.0f / (r * r);                    // precompute inverse radius^2
    v[14] = 0.0f;
    v[15] = 0.0f;

    float4* dst = (float4*)(rec + (size_t)n * RECW);
#pragma unroll
    for (int q = 0; q < 4; ++q)
        dst[q] = make_float4(v[4 * q + 0], v[4 * q + 1], v[4 * q + 2], v[4 * q + 3]);
}

#if USE_TDM
// ---------------------------------------------------------------------------
// Build a 2D TDM descriptor (D#) and issue TENSOR_LOAD_TO_LDS.
// Tile: tile_y rows of tile_x 4-byte elements, row stride = row_stride elems.
// D# bit layout per CDNA5 ISA ch.8 (group0 128b, group1 256b).
// ---------------------------------------------------------------------------
__device__ __forceinline__ void tdm_load_tile_2d(uint32_t lds_byte_off,
                                                 const void* gsrc,
                                                 uint32_t tile_x, uint32_t tile_y,
                                                 uint64_t tensor_x, uint64_t row_stride)
{
    const uint64_t ga = (uint64_t)(uintptr_t)gsrc;
    u32x4 g0;
    g0[0] = 0x1u;                                   // count=1 (valid), user mode
    g0[1] = lds_byte_off;                           // D#.lds_addr
    g0[2] = (uint32_t)ga;                           // global_addr[31:0]  (bits 95:64)
    g0[3] = (uint32_t)((ga >> 32) & 0x01FFFFFFu)    // global_addr[56:32]
          | (2u << 30);                             // D#.type = 2 ("image")
    i32x8 g1;
    g1[0] = (int)(2u << 16);                        // data_size = 2 -> 4 bytes; mask=0
    g1[1] = (int)(((uint32_t)tensor_x & 0xFFFFu) << 16);        // tensor_dim0[15:0]
    g1[2] = (int)((((uint32_t)(tensor_x >> 16)) & 0xFFFFu)      // tensor_dim0[31:16]
          | ((tile_y & 0xFFFFu) << 16));                        // tensor_dim1[15:0]
    g1[3] = (int)(((tile_y >> 16) & 0xFFFFu)                    // tensor_dim1[31:16]
          | ((tile_x & 0xFFFFu) << 16));                        // tile_dim0
    g1[4] = (int)(tile_y & 0xFFFFu);                // tile_dim1 (tile_dim2 = 0)
    g1[5] = (int)(uint32_t)row_stride;              // tensor_dim0_stride[31:0]
    g1[6] = (int)((uint32_t)(row_stride >> 32) & 0xFFFFu); // stride[47:32]
    g1[7] = 0;                                      // tensor_dim1_stride = 0 (2D tile)
    i32x4 gz4 = {0, 0, 0, 0};
#if defined(__clang_major__) && (__clang_major__ >= 23)
    i32x8 gz8 = {0, 0, 0, 0, 0, 0, 0, 0};
    __builtin_amdgcn_tensor_load_to_lds(g0, g1, gz4, gz4, gz8, 0);
#else
    __builtin_amdgcn_tensor_load_to_lds(g0, g1, gz4, gz4, 0);
#endif
}
#endif // USE_TDM

// ---------------------------------------------------------------------------
// Kernel 2: alpha compositing. One thread per pixel; a 256-thread block owns a
// 256-pixel tile. All 24 K-slices of idx/dists2 for the tile are staged into
// LDS by the CDNA5 Tensor Data Mover (one 2D descriptor per stream, issued by
// wave 0, tracked by TENSORcnt), then each thread runs the serial
// transmittance scan with 64B record gathers.
// ---------------------------------------------------------------------------
__global__ __launch_bounds__(256)
void composite_kernel(const float* __restrict__ d2_g,   // [K, H*W]
                      const int*   __restrict__ idx_g,  // [K, H*W]
                      const float* __restrict__ rec,    // [N, 16]
                      float* __restrict__ out)          // [C, H*W]
{
    __shared__ int   s_idx[KK][TPB];   // 24 KB
    __shared__ float s_d2 [KK][TPB];   // 24 KB

    const int tid = threadIdx.x;
    const int p0  = blockIdx.x * TPB;  // first pixel of this tile

#if USE_TDM
    // ---- Stage via TDM: two 2D tiles (24 rows x 1KB, row stride HW*4B). ----
    if (tid < 32) {                    // wave 0 issues the DMA descriptors
        tdm_load_tile_2d((uint32_t)(uintptr_t)&s_idx[0][0], idx_g + p0,
                         TPB, KK, HWPX, HWPX);
        tdm_load_tile_2d((uint32_t)(uintptr_t)&s_d2[0][0],  d2_g + p0,
                         TPB, KK, HWPX, HWPX);
        __builtin_amdgcn_s_wait_tensorcnt(0);
    }
    __syncthreads();
#else
    // ---- Fallback: per-lane CDNA5 async global->LDS copies (ASYNCcnt). ----
#pragma unroll
    for (int j = 0; j < 6; ++j) {
        int chunk = j * 256 + tid;          // 0..1535 int4 chunks per stream
        int k     = chunk >> 6;             // 64 int4 chunks per slice
        int off   = (chunk & 63) << 2;      // int offset within slice

        const int*   gi = idx_g + (size_t)k * HWPX + p0 + off;
        const float* gf = d2_g  + (size_t)k * HWPX + p0 + off;
        uint32_t li = (uint32_t)(uintptr_t)(&s_idx[0][0] + chunk * 4);
        uint32_t lf = (uint32_t)(uintptr_t)(&s_d2 [0][0] + chunk * 4);
        asm volatile("global_load_async_to_lds_b128 %0, %1, off"
                     :: "v"(li), "v"(gi) : "memory");
        asm volatile("global_load_async_to_lds_b128 %0, %1, off"
                     :: "v"(lf), "v"(gf) : "memory");
    }
    asm volatile("s_wait_asynccnt 0" ::: "memory");
    __syncthreads();
#endif

    // ---- Compute: serial transmittance scan over K for pixel p. ----
    const int p = p0 + tid;

    float T = 1.0f;            // running transmittance prod_{j<k}(1 - alpha_j)
    float acc[CF];
#pragma unroll
    for (int c = 0; c < CF; ++c) acc[c] = 0.0f;

#pragma unroll
    for (int k = 0; k < KK; ++k) {
        int   id = s_idx[k][tid];
        float d2 = s_d2 [k][tid];

        if (id >= 0) {
            const float4* r4 = (const float4*)(rec + (size_t)id * RECW);
            float4 a = r4[0];
            float4 b = r4[1];
            float4 c = r4[2];
            float4 d = r4[3];

            float alpha = 1.0f - d2 * d.y;   // d.y = 1/r^2
            float w     = alpha * T;         // weight = alpha * prev_trans
            T -= w;                          // T *= (1 - alpha)

            acc[ 0] += w * a.x;  acc[ 1] += w * a.y;
            acc[ 2] += w * a.z;  acc[ 3] += w * a.w;
            acc[ 4] += w * b.x;  acc[ 5] += w * b.y;
            acc[ 6] += w * b.z;  acc[ 7] += w * b.w;
            acc[ 8] += w * c.x;  acc[ 9] += w * c.y;
            acc[10] += w * c.z;  acc[11] += w * c.w;
            acc[12] += w * d.x;
        }
        // masked fragment: alpha = 0 -> weight 0, T unchanged (matches ref)
    }

    // Streamed output: NT stores keep L2 capacity for the gather table.
#pragma unroll
    for (int c = 0; c < CF; ++c)
        __builtin_nontemporal_store(acc[c], &out[(size_t)c * HWPX + p]);
}

// ---------------------------------------------------------------------------
// Launch: inputs in setup_inputs() order: dists2, point_radius, features, idx.
// d_ws holds the packed point records (200000 * 16 * 4 B = 12.8 MB).
// ---------------------------------------------------------------------------
extern "C" void kernel_launch(void* const* d_in, const int* in_sizes, int n_in,
                              void* d_out, int out_size, void* d_ws, size_t ws_size,
                              hipStream_t stream)
{
    const float* dists2 = (const float*)d_in[0];   // [1,24,512,512] f32
    const float* radius = (const float*)d_in[1];   // [200000] f32
    const float* feats  = (const float*)d_in[2];   // [13,200000] f32
    const int*   idx    = (const int*)  d_in[3];   // [1,24,512,512] i32
    float*       out    = (float*)d_out;           // [1,13,512,512] f32
    float*       rec    = (float*)d_ws;            // [200000,16] scratch

    (void)in_sizes; (void)n_in; (void)out_size; (void)ws_size;

    pack_points_kernel<<<(NPTS + 255) / 256, 256, 0, stream>>>(feats, radius, rec);
    composite_kernel<<<HWPX / TPB, TPB, 0, stream>>>(dists2, idx, rec, out);
}